// HmmTrajNet2_10660108829198
// MI455X (gfx1250) — compile-verified
//
#include <hip/hip_runtime.h>
#include <hip/hip_bf16.h>

#define NEGF (-1.0e30f)

typedef float v2f __attribute__((ext_vector_type(2)));
typedef float v8f __attribute__((ext_vector_type(8)));

#define TT 512
#define BB 64
#define AA 16

__device__ __forceinline__ float lse2(float a, float b) {
    float m = fmaxf(a, b);
    return m + __logf(__expf(a - m) + __expf(b - m));
}

// ---------- K1a: gather action logps: a[t,b] = alf[t,b,actions[t]] ----------
__global__ void k_gather(const float* __restrict__ alf, const int* __restrict__ actions,
                         float* __restrict__ a) {
    int idx = blockIdx.x * blockDim.x + threadIdx.x;
    if (idx >= TT * BB) return;
    int t = idx >> 6, b = idx & 63;
    a[idx] = alf[(t * BB + b) * AA + actions[t]];
}

// ---------- K1b: per-b prefix sums -> U[b,j], W[t,b] (64 indep threads) ----------
__global__ void k_prefix(const float* __restrict__ a, const float* __restrict__ stop,
                         const float* __restrict__ start,
                         float* __restrict__ W, float* __restrict__ U) {
    int b = threadIdx.x;               // 0..63
    float A0 = 0.0f;                   // A0[0]
    for (int t = 1; t <= TT; ++t) {
        float st0 = stop[((t) * BB + b) * 2 + 0];
        float st1 = stop[((t) * BB + b) * 2 + 1];
        float at  = a[(t - 1) * BB + b];
        W[(t - 1) * BB + b] = A0 + at + st1;           // W[t,b] = A0[t-1]+a[t-1]+st1[t]
        A0 += at + st0;                                 // A0[t]
        U[b * TT + (t - 1)] = start[(t - 1) * BB + b] + at + st0 - A0; // U[b,j], j=t-1
    }
}

// ---------- K1c: row/col maxes + exponentials for stable exp-space GEMM ----------
__global__ void k_maxexp(const float* __restrict__ W, const float* __restrict__ U,
                         float* __restrict__ eW, float* __restrict__ eU,
                         float* __restrict__ mW, float* __restrict__ mU) {
    __shared__ float red[BB];
    int b = threadIdx.x;
    int blk = blockIdx.x;              // 0..511: W row; 512..1023: U col
    float x = (blk < TT) ? W[blk * BB + b] : U[b * TT + (blk - TT)];
    red[b] = x; __syncthreads();
    for (int off = 32; off; off >>= 1) {
        if (b < off) red[b] = fmaxf(red[b], red[b + off]);
        __syncthreads();
    }
    float m = red[0];
    if (blk < TT) { if (b == 0) mW[blk] = m; eW[blk * BB + b] = __expf(x - m); }
    else { int j = blk - TT; if (b == 0) mU[j] = m; eU[b * TT + j] = __expf(x - m); }
}

// ---------- K2: WMMA log-space GEMM: V[t,j] = mW+mU+log(sum_b eW[t,b]*eU[b,j]) ----------
__global__ void k_logmm(const float* __restrict__ eW, const float* __restrict__ eU,
                        const float* __restrict__ mW, const float* __restrict__ mU,
                        float* __restrict__ V) {
    int wave = blockIdx.x * (blockDim.x >> 5) + (threadIdx.x >> 5);
    int lane = threadIdx.x & 31;
    const int tilesN = TT / 16;                 // 32
    int tm = (wave / tilesN) * 16;              // row tile base (t-1 axis, M)
    int tn = (wave % tilesN) * 16;              // col tile base (j axis, N)
    int half = lane >> 4;                       // 0 or 1
    int l16 = lane & 15;
    v8f acc = {};
#pragma unroll
    for (int k0 = 0; k0 < BB; k0 += 4) {
        int ka = k0 + half * 2;
        // A fragment 16x4 (ISA: VGPR0 holds K=0|2, VGPR1 holds K=1|3 by lane half)
        v2f af;
        af.x = eW[(tm + l16) * BB + ka];
        af.y = eW[(tm + l16) * BB + ka + 1];
        // B fragment 4x16 (row K striped across lanes, mirrored K split)
        v2f bf;
        bf.x = eU[(ka) * TT + tn + l16];
        bf.y = eU[(ka + 1) * TT + tn + l16];
        acc = __builtin_amdgcn_wmma_f32_16x16x4_f32(false, af, false, bf,
                                                    (short)0, acc, false, false);
    }
    // C/D layout: VGPR r -> (M=r, N=lane) lanes 0-15 ; (M=r+8, N=lane-16) lanes 16-31
#pragma unroll
    for (int r = 0; r < 8; ++r) {
        int row = tm + r + half * 8;            // t-1 index
        int col = tn + l16;                     // j
        float s = acc[r];
        V[row * TT + col] = mW[row] + mU[col] + __logf(fmaxf(s, 1e-38f));
    }
}

// ---------- K3: sequential C-scan (single wave, shuffle reductions) ----------
__global__ void k_cscan(const float* __restrict__ V, float* __restrict__ Cout,
                        float* __restrict__ tl) {
    __shared__ float C[TT + 2];
    int lane = threadIdx.x;
    if (lane == 0) { C[0] = NEGF; C[1] = 0.0f; }
    __syncthreads();
    for (int t = 2; t <= TT; ++t) {
        int n = t - 1;                          // valid j = 0..t-2
        const float* row = V + (size_t)(t - 2) * TT;   // V[t-1]
        float m = NEGF;
        for (int j = lane; j < n; j += 32) m = fmaxf(m, C[j + 1] + row[j]);
        for (int i = 16; i; i >>= 1) m = fmaxf(m, __shfl_xor(m, i, 32));
        float s = 0.0f;
        for (int j = lane; j < n; j += 32) s += __expf(C[j + 1] + row[j] - m);
        for (int i = 16; i; i >>= 1) s += __shfl_xor(s, i, 32);
        if (lane == 0) C[t] = m + __logf(fmaxf(s, 1e-38f));
        __syncthreads();
    }
    // total_logp = logsumexp_j (C[j+1] + V[T, j])
    const float* row = V + (size_t)(TT - 1) * TT;
    float m = NEGF;
    for (int j = lane; j < TT; j += 32) m = fmaxf(m, C[j + 1] + row[j]);
    for (int i = 16; i; i >>= 1) m = fmaxf(m, __shfl_xor(m, i, 32));
    float s = 0.0f;
    for (int j = lane; j < TT; j += 32) s += __expf(C[j + 1] + row[j] - m);
    for (int i = 16; i; i >>= 1) s += __shfl_xor(s, i, 32);
    if (lane == 0) tl[0] = m + __logf(fmaxf(s, 1e-38f));
    for (int j = lane; j <= TT; j += 32) Cout[j] = C[j];
}

// ---------- K4: backward g-scan (single wave, 2 b-values per lane) ----------
__global__ void k_bscan(const float* __restrict__ a, const float* __restrict__ stop,
                        const float* __restrict__ start, float* __restrict__ G1) {
    int lane = threadIdx.x;
    int b0 = lane, b1 = lane + 32;
    float gc0a = NEGF, gc1a = 0.0f;     // g[b0][0], g[b0][1]
    float gc0b = NEGF, gc1b = 0.0f;     // g[b1][0], g[b1][1]
    if (lane == 0) G1[TT] = 0.0f;
    for (int t = TT - 1; t >= 0; --t) {
        float sa0 = stop[((t + 1) * BB + b0) * 2 + 0];
        float sa1 = stop[((t + 1) * BB + b0) * 2 + 1];
        float sb0 = stop[((t + 1) * BB + b1) * 2 + 0];
        float sb1 = stop[((t + 1) * BB + b1) * 2 + 1];
        float la = lse2(sa0 + gc0a, sa1 + gc1a);
        float lb = lse2(sb0 + gc0b, sb1 + gc1b);
        float n0a = a[t * BB + b0] + la;
        float n0b = a[t * BB + b1] + lb;
        float ta = start[t * BB + b0] + n0a;
        float tb = start[t * BB + b1] + n0b;
        float m = fmaxf(ta, tb);
        for (int i = 16; i; i >>= 1) m = fmaxf(m, __shfl_xor(m, i, 32));
        float s = __expf(ta - m) + __expf(tb - m);
        for (int i = 16; i; i >>= 1) s += __shfl_xor(s, i, 32);
        float g1 = m + __logf(s);
        if (t >= 1 && lane == 0) G1[t] = g1;
        gc0a = n0a; gc0b = n0b; gc1a = g1; gc1b = g1;
    }
}

// ---------- K5: streaming (max,sum) over causal penalties, one block per t ----------
__global__ void k_cc(const float* __restrict__ P, const float* __restrict__ C,
                     const float* __restrict__ U, const float* __restrict__ W,
                     const float* __restrict__ G1, float* __restrict__ pairs) {
    int t = blockIdx.x + 1;             // 1..511
    int tid = threadIdx.x;              // 0..255
    float base = G1[t + 1];
    float m = NEGF, s = 0.0f;
    int nelem = t * BB;                 // j in [0,t), b in [0,64)
    for (int e = tid; e < nelem; e += 256) {
        int j = e >> 6, b = e & 63;
        float x = C[j + 1] + U[b * TT + j] + W[(t - 1) * BB + b] + base
                + P[((size_t)t * (TT + 1) + j) * BB + b];
        if (x > m) { s = s * __expf(m - x) + 1.0f; m = x; }
        else s += __expf(x - m);
    }
    __shared__ float sm[256], ss[256];
    sm[tid] = m; ss[tid] = s; __syncthreads();
    for (int off = 128; off; off >>= 1) {
        if (tid < off) {
            float m2 = sm[tid + off], s2 = ss[tid + off];
            float M = fmaxf(sm[tid], m2);
            ss[tid] = ss[tid] * __expf(sm[tid] - M) + s2 * __expf(m2 - M);
            sm[tid] = M;
        }
        __syncthreads();
    }
    if (tid == 0) { pairs[blockIdx.x * 2] = sm[0]; pairs[blockIdx.x * 2 + 1] = ss[0]; }
}

// ---------- K6: combine pairs, emit cc_loss - total_logp ----------
__global__ void k_final(const float* __restrict__ pairs, const float* __restrict__ tl,
                        float* __restrict__ out) {
    __shared__ float sm[256], ss[256];
    int tid = threadIdx.x;
    float m = NEGF, s = 0.0f;
    for (int i = tid; i < TT - 1; i += 256) {
        float m2 = pairs[2 * i], s2 = pairs[2 * i + 1];
        float M = fmaxf(m, m2);
        s = s * __expf(m - M) + s2 * __expf(m2 - M);
        m = M;
    }
    sm[tid] = m; ss[tid] = s; __syncthreads();
    for (int off = 128; off; off >>= 1) {
        if (tid < off) {
            float m2 = sm[tid + off], s2 = ss[tid + off];
            float M = fmaxf(sm[tid], m2);
            ss[tid] = ss[tid] * __expf(sm[tid] - M) + s2 * __expf(m2 - M);
            sm[tid] = M;
        }
        __syncthreads();
    }
    if (tid == 0) out[0] = (sm[0] + __logf(fmaxf(ss[0], 1e-38f))) - tl[0];
}

extern "C" void kernel_launch(void* const* d_in, const int* in_sizes, int n_in,
                              void* d_out, int out_size, void* d_ws, size_t ws_size,
                              hipStream_t stream) {
    const float* alf     = (const float*)d_in[0];  // (513,64,16)
    const float* stop    = (const float*)d_in[1];  // (513,64,2)
    const float* start   = (const float*)d_in[2];  // (513,64)
    const float* pen     = (const float*)d_in[3];  // (513,513,64)
    const int*   actions = (const int*)d_in[4];    // (512,)
    float* out = (float*)d_out;

    float* ws   = (float*)d_ws;
    float* a    = ws;                  // 512*64
    float* Wb   = a  + TT * BB;        // 512*64
    float* Ub   = Wb + TT * BB;        // 64*512
    float* eW   = Ub + TT * BB;        // 512*64
    float* eU   = eW + TT * BB;        // 64*512
    float* mW   = eU + TT * BB;        // 512
    float* mU   = mW + TT;             // 512
    float* Vb   = mU + TT;             // 512*512
    float* Cb   = Vb + TT * TT;        // 513
    float* G1   = Cb + (TT + 1);       // 513
    float* tl   = G1 + (TT + 1);       // 1
    float* pairs= tl + 1;              // 2*(511)

    k_gather<<<dim3((TT * BB + 255) / 256), dim3(256), 0, stream>>>(alf, actions, a);
    k_prefix<<<dim3(1), dim3(BB), 0, stream>>>(a, stop, start, Wb, Ub);
    k_maxexp<<<dim3(2 * TT), dim3(BB), 0, stream>>>(Wb, Ub, eW, eU, mW, mU);
    // 32x32 tiles of 16x16, one wave each; 4 waves per block
    k_logmm<<<dim3((TT / 16) * (TT / 16) / 4), dim3(128), 0, stream>>>(eW, eU, mW, mU, Vb);
    k_cscan<<<dim3(1), dim3(32), 0, stream>>>(Vb, Cb, tl);
    k_bscan<<<dim3(1), dim3(32), 0, stream>>>(a, stop, start, G1);
    k_cc<<<dim3(TT - 1), dim3(256), 0, stream>>>(pen, Cb, Ub, Wb, G1, pairs);
    k_final<<<dim3(1), dim3(256), 0, stream>>>(pairs, tl, out);
}